// OhemCELoss_6158983102532
// MI455X (gfx1250) — compile-verified
//
#include <hip/hip_runtime.h>

// ---------------------------------------------------------------------------
// OHEM CE loss, MI455X (gfx1250, wave32).
// out[p] = 0.5*CE(score0)[p] + 0.5*ohem(score1)   (ohem is a scalar)
// Memory-bound: ~370 MB total traffic -> ~16 us at 23.3 TB/s. No GEMM content,
// so no WMMA; the win is one fused streaming pass (b128 loads, 4 pixels per
// thread) + histogram-based selection instead of a global sort, plus the
// gfx1250 async-to-LDS path (ASYNCcnt) in the reduction.
//
// Workspace layout (needs ~12.6 MB):
//   [0       .. 4MB)     uint32 hist[2^20]
//   [4MB     .. 12.4MB)  float  pred[N]
//   [12.4MB  .. +32B)    float  scal[8]: [0]=threshold [1]=ohem [2]=n_valid(u32)
//   then float psum[256], pcnt[256]
// ---------------------------------------------------------------------------

#define C_K 19
#define HW_K (512 * 512)
#define N_K (8 * HW_K)
#define NBINS (1 << 20)
#define IGNORE_K 255
#define THRESH_K 0.7f
#define MIN_KEPT_K 100000
#define RED_BLOCKS 256
#define TPB 256

// ---- gfx1250 async-to-LDS (existence proven by round-1 diagnostics) -------
#if defined(__HIP_DEVICE_COMPILE__) &&                                         \
    __has_builtin(__builtin_amdgcn_global_load_async_to_lds_b32) &&            \
    __has_builtin(__builtin_amdgcn_s_wait_asynccnt)
#define USE_ASYNC_LDS 1
#else
#define USE_ASYNC_LDS 0
#endif

// Builtin signature (from hipcc diagnostic, round 1):
//   void __builtin_amdgcn_global_load_async_to_lds_b32(
//       as1 int* gsrc, as3 int* ldst, imm int offset, imm int cpol)
typedef __attribute__((address_space(1))) int as1_int;
typedef __attribute__((address_space(3))) int as3_int;

// LDS generic pointers carry the LDS byte offset in the low 32 bits
// (ISA 10.2) -> integer truncation yields a valid
// 32-bit AS3 pointer. Global generic pointers are bitwise-identical to AS1.
__device__ __forceinline__ as3_int* lds_cast(void* p) {
  return (as3_int*)(unsigned int)(unsigned long long)p;
}
__device__ __forceinline__ as1_int* glob_cast(const void* p) {
  return (as1_int*)(unsigned long long)p;
}

// compile-time-constant component select (folds under #pragma unroll)
__device__ __forceinline__ float comp(const float4& v, int j) {
  return (j == 0) ? v.x : (j == 1) ? v.y : (j == 2) ? v.z : v.w;
}

// ---------------------------------------------------------------------------
// Kernel 1: zero histogram + scalar/partial area
// ---------------------------------------------------------------------------
__global__ void __launch_bounds__(TPB)
zero_kernel(uint32_t* __restrict__ hist, uint32_t* __restrict__ tail) {
  int i = blockIdx.x * TPB + threadIdx.x;
  if (i < NBINS) {
    hist[i] = 0u;
  } else if (i - NBINS < 8 + 2 * RED_BLOCKS) {
    tail[i - NBINS] = 0u;
  }
}

// ---------------------------------------------------------------------------
// Kernel 2: fused main pass, 4 pixels per thread, b128 loads/stores.
//  - log-softmax over C=19 of score0  -> out = 0.5 * CE
//  - softmax prob of target class of score1 -> pred[] (2.0 sentinel if ignored)
//  - histogram of pred (valid only), wave32 ballot count of valid pixels
// Channel loads are float4, stride-1 across the wave (512B per wave per load),
// channel stride HW*4 bytes. score1 lines prefetched (global_prefetch_b8)
// while the score0 softmax executes.
// ---------------------------------------------------------------------------
__global__ void __launch_bounds__(TPB)
main_kernel(const float* __restrict__ s0g, const float* __restrict__ s1g,
            const int* __restrict__ tg, float* __restrict__ out,
            float* __restrict__ pred, uint32_t* __restrict__ hist,
            uint32_t* __restrict__ nvalid) {
  const int tid4 = blockIdx.x * TPB + threadIdx.x;  // one thread = 4 pixels
  const int p4 = tid4 * 4;
  const int b = p4 >> 18;                           // p / HW
  const int hw = p4 & (HW_K - 1);                   // p % HW (multiple of 4)
  const size_t base = (size_t)b * (size_t)(C_K * HW_K) + (size_t)hw;
  const float4* s0 = (const float4*)(s0g + base);
  const float4* s1 = (const float4*)(s1g + base);
  const size_t CH4 = (size_t)(HW_K / 4);            // channel stride in float4

  const int4 t4 = ((const int4*)tg)[tid4];
  int tcs[4];
  bool val[4];
  tcs[0] = t4.x; tcs[1] = t4.y; tcs[2] = t4.z; tcs[3] = t4.w;
#pragma unroll
  for (int j = 0; j < 4; ++j) {
    val[j] = (tcs[j] != IGNORE_K);
    tcs[j] = val[j] ? tcs[j] : 0;
  }

  // Warm score1's lines into cache while we crunch score0 (gfx1250 prefetch).
#pragma unroll
  for (int c = 0; c < C_K; ++c)
    __builtin_prefetch((const void*)(s1 + (size_t)c * CH4), 0, 3);

  float4 xv[C_K];

  // ---- branch 0: per-pixel CE of score0 ----
#pragma unroll
  for (int c = 0; c < C_K; ++c) xv[c] = s0[(size_t)c * CH4];

  float ce[4];
#pragma unroll
  for (int j = 0; j < 4; ++j) {
    float m = comp(xv[0], j);
#pragma unroll
    for (int c = 1; c < C_K; ++c) m = fmaxf(m, comp(xv[c], j));
    float se = 0.0f, xt = comp(xv[0], j);
#pragma unroll
    for (int c = 0; c < C_K; ++c) {
      float v = comp(xv[c], j);
      se += __expf(v - m);
      xt = (c == tcs[j]) ? v : xt;
    }
    float lse = m + __logf(se);
    ce[j] = val[j] ? (lse - xt) : 0.0f;
  }
  ((float4*)out)[tid4] =
      make_float4(0.5f * ce[0], 0.5f * ce[1], 0.5f * ce[2], 0.5f * ce[3]);

  // ---- branch 1: pred = softmax(score1)[target] ----
#pragma unroll
  for (int c = 0; c < C_K; ++c) xv[c] = s1[(size_t)c * CH4];

  float pr[4];
#pragma unroll
  for (int j = 0; j < 4; ++j) {
    float m = comp(xv[0], j);
#pragma unroll
    for (int c = 1; c < C_K; ++c) m = fmaxf(m, comp(xv[c], j));
    float se = 0.0f, xt = comp(xv[0], j);
#pragma unroll
    for (int c = 0; c < C_K; ++c) {
      float v = comp(xv[c], j);
      se += __expf(v - m);
      xt = (c == tcs[j]) ? v : xt;
    }
    float lse = m + __logf(se);
    pr[j] = val[j] ? __expf(xt - lse) : 2.0f;   // sentinel > any threshold
  }
  ((float4*)pred)[tid4] = make_float4(pr[0], pr[1], pr[2], pr[3]);

#pragma unroll
  for (int j = 0; j < 4; ++j) {
    if (val[j]) {
      int bin = (int)(pr[j] * (float)NBINS);
      bin = bin < 0 ? 0 : bin;
      bin = bin > (NBINS - 1) ? (NBINS - 1) : bin;
      atomicAdd(&hist[bin], 1u);   // low contention: 2.1M adds over 1M bins
    }
  }

  // valid-pixel count: wave32 ballots -> one atomic per wave
  unsigned cnt = 0;
#pragma unroll
  for (int j = 0; j < 4; ++j) {
    unsigned long long bal = __ballot(val[j] ? 1 : 0);
    if ((threadIdx.x & 31) == 0) cnt += (unsigned)__popcll(bal);
  }
  if ((threadIdx.x & 31) == 0 && cnt) atomicAdd(nvalid, cnt);
}

// ---------------------------------------------------------------------------
// Kernel 3: single-block rank selection over the 2^20-bin histogram.
// Finds the bin containing rank idx = min(MIN_KEPT, n_valid-1); threshold =
// max(bin upper edge, 0.7). scal[0] = threshold.
// ---------------------------------------------------------------------------
__global__ void __launch_bounds__(1024)
select_kernel(const uint32_t* __restrict__ hist,
              const uint32_t* __restrict__ nvalid_p,
              float* __restrict__ scal) {
  __shared__ uint32_t ps[1024];
  const int tid = threadIdx.x;
  const int CH = NBINS / 1024;
  const uint32_t* hbase = hist + tid * CH;

  uint32_t s = 0;
  for (int i = 0; i < CH; ++i) s += hbase[i];
  ps[tid] = s;
  __syncthreads();

  // inclusive Hillis-Steele scan over the 1024 chunk sums
  for (int off = 1; off < 1024; off <<= 1) {
    uint32_t v = (tid >= off) ? ps[tid - off] : 0u;
    __syncthreads();
    ps[tid] += v;
    __syncthreads();
  }

  uint32_t nv = *nvalid_p;
  long long idx = (long long)MIN_KEPT_K;
  if ((long long)nv - 1 < idx) idx = (long long)nv - 1;
  if (idx < 0) {               // no valid pixels: fall back to fixed threshold
    if (tid == 0) scal[0] = THRESH_K;
    return;
  }
  uint32_t need = (uint32_t)(idx + 1);
  uint32_t before = (tid > 0) ? ps[tid - 1] : 0u;
  uint32_t incl = ps[tid];
  if (before < need && incl >= need) {   // exactly one thread matches
    uint32_t cum = before;
    int bin = tid * CH + CH - 1;
    for (int i = 0; i < CH; ++i) {
      cum += hbase[i];
      if (cum >= need) { bin = tid * CH + i; break; }
    }
    float minv = (float)(bin + 1) * (1.0f / (float)NBINS);  // bin upper edge
    scal[0] = fmaxf(minv, THRESH_K);
  }
}

// ---------------------------------------------------------------------------
// Kernel 4: grid-stride masked reduction of (-log(pred), 1) for pred < thr.
// Tiles of pred are staged LDS-side via the gfx1250 async-to-LDS path
// (ASYNCcnt + s_wait_asynccnt). Deterministic: tree reduce to 256 per-block
// partials, no float atomics anywhere.
// ---------------------------------------------------------------------------
__global__ void __launch_bounds__(TPB)
reduce_kernel(const float* __restrict__ pred, const float* __restrict__ scal,
              float* __restrict__ psum, float* __restrict__ pcnt) {
  const float thr = scal[0];
  __shared__ float tile[TPB];
  float lsum = 0.0f, lcnt = 0.0f;
  const int stride = RED_BLOCKS * TPB;
  for (int base = blockIdx.x * TPB; base < N_K; base += stride) {
    int i = base + threadIdx.x;
#if USE_ASYNC_LDS
    __builtin_amdgcn_global_load_async_to_lds_b32(
        glob_cast(pred + i), lds_cast(&tile[threadIdx.x]), 0, 0);
    __builtin_amdgcn_s_wait_asynccnt(0);
    __syncthreads();
    float pr = tile[threadIdx.x];
    __syncthreads();
#else
    tile[threadIdx.x] = pred[i];
    __syncthreads();
    float pr = tile[threadIdx.x];
    __syncthreads();
#endif
    if (pr < thr) {            // ignored pixels carry sentinel 2.0 -> excluded
      lsum -= __logf(pr);      // CE loss == -log(pred) exactly
      lcnt += 1.0f;
    }
  }
  __shared__ float rs[TPB], rc[TPB];
  rs[threadIdx.x] = lsum;
  rc[threadIdx.x] = lcnt;
  __syncthreads();
  for (int off = TPB / 2; off > 0; off >>= 1) {
    if (threadIdx.x < off) {
      rs[threadIdx.x] += rs[threadIdx.x + off];
      rc[threadIdx.x] += rc[threadIdx.x + off];
    }
    __syncthreads();
  }
  if (threadIdx.x == 0) {
    psum[blockIdx.x] = rs[0];
    pcnt[blockIdx.x] = rc[0];
  }
}

// ---------------------------------------------------------------------------
// Kernel 5: finalize ohem scalar. scal[1] = sum(kept losses) / count(kept).
// ---------------------------------------------------------------------------
__global__ void __launch_bounds__(RED_BLOCKS)
finalize_kernel(const float* __restrict__ psum, const float* __restrict__ pcnt,
                float* __restrict__ scal) {
  __shared__ float rs[RED_BLOCKS], rc[RED_BLOCKS];
  int tid = threadIdx.x;
  rs[tid] = psum[tid];
  rc[tid] = pcnt[tid];
  __syncthreads();
  for (int off = RED_BLOCKS / 2; off > 0; off >>= 1) {
    if (tid < off) {
      rs[tid] += rs[tid + off];
      rc[tid] += rc[tid + off];
    }
    __syncthreads();
  }
  if (tid == 0) scal[1] = rs[0] / rc[0];
}

// ---------------------------------------------------------------------------
// Kernel 6: broadcast-add the ohem scalar (b128): out += 0.5 * ohem.
// ---------------------------------------------------------------------------
__global__ void __launch_bounds__(TPB)
broadcast_kernel(float4* __restrict__ out4, const float* __restrict__ scal) {
  float add = 0.5f * scal[1];
  int i = blockIdx.x * TPB + threadIdx.x;
  float4 v = out4[i];
  v.x += add; v.y += add; v.z += add; v.w += add;
  out4[i] = v;
}

// ---------------------------------------------------------------------------
extern "C" void kernel_launch(void* const* d_in, const int* in_sizes, int n_in,
                              void* d_out, int out_size, void* d_ws,
                              size_t ws_size, hipStream_t stream) {
  (void)in_sizes; (void)n_in; (void)out_size; (void)ws_size;
  const float* score0 = (const float*)d_in[0];
  const float* score1 = (const float*)d_in[1];
  const int* target = (const int*)d_in[2];
  float* out = (float*)d_out;

  char* ws = (char*)d_ws;
  uint32_t* hist = (uint32_t*)ws;
  float* pred = (float*)(ws + (size_t)NBINS * 4);
  float* scal = (float*)(ws + (size_t)NBINS * 4 + (size_t)N_K * 4);
  float* psum = scal + 8;
  float* pcnt = psum + RED_BLOCKS;
  uint32_t* nvalid = (uint32_t*)(scal + 2);

  const int zero_words = NBINS + 8 + 2 * RED_BLOCKS;
  zero_kernel<<<(zero_words + TPB - 1) / TPB, TPB, 0, stream>>>(
      hist, (uint32_t*)scal);
  main_kernel<<<(N_K / 4) / TPB, TPB, 0, stream>>>(score0, score1, target, out,
                                                   pred, hist, nvalid);
  select_kernel<<<1, 1024, 0, stream>>>(hist, nvalid, scal);
  reduce_kernel<<<RED_BLOCKS, TPB, 0, stream>>>(pred, scal, psum, pcnt);
  finalize_kernel<<<1, RED_BLOCKS, 0, stream>>>(psum, pcnt, scal);
  broadcast_kernel<<<(N_K / 4) / TPB, TPB, 0, stream>>>((float4*)out, scal);
}